// GESelfAttentionBlock_55362128445887
// MI455X (gfx1250) — compile-verified
//
#include <hip/hip_runtime.h>

// Problem dims (fixed by reference)
#define VNUM 20000
#define NEI  12
#define CH   16
#define NN   8
#define HH   4

typedef __attribute__((ext_vector_type(2))) float v2f;
typedef __attribute__((ext_vector_type(4))) float flt4;
typedef __attribute__((ext_vector_type(8))) float v8f;

// Workspace layout (float offsets)
#define WS_WQ    0            // [32][128]
#define WS_WK    4096         // [32][128]
#define WS_WV    8192         // [4][16][8][48]
#define WS_WMT   32768        // [128][512]  (WMt[p][k] = WM[o,hc,i,j], p=(o,i), k=(hc,j))
#define WS_KBUF  98304        // [V][12][32]
#define WS_ATTN  7778304      // [V][12][4]
#define WS_HEAD  8738304      // [V][512]

// ---------------------------------------------------------------------------
// Kernel 1: expand all basis-combined weight matrices into workspace.
// ---------------------------------------------------------------------------
__global__ void k_weights(const float* __restrict__ rb,   // [8][8][8]
                          const float* __restrict__ vb1,  // [8][2][8][8]
                          const float* __restrict__ vb2,  // [8][3][8][8]
                          const float* __restrict__ qc,   // [4][16][8]
                          const float* __restrict__ kc,
                          const float* __restrict__ v0p,
                          const float* __restrict__ v1p,
                          const float* __restrict__ v2p,
                          const float* __restrict__ wm,   // [16][64][8]
                          float* __restrict__ ws) {
  const int total = 4096 + 4096 + 24576 + 65536;  // 98304
  for (int idx = blockIdx.x * blockDim.x + threadIdx.x; idx < total;
       idx += gridDim.x * blockDim.x) {
    if (idx < 8192) {                    // WQ / WK rows: [hi][cj]
      int t = idx & 4095;
      const float* coef = (idx < 4096) ? qc : kc;
      int hi = t >> 7, cj = t & 127;
      int h = hi >> 3, i = hi & 7, c = cj >> 3, j = cj & 7;
      float s = 0.f;
#pragma unroll
      for (int b = 0; b < 8; ++b)
        s += coef[(h * 16 + c) * 8 + b] * rb[(b * 8 + i) * 8 + j];
      ws[idx] = s;
    } else if (idx < 32768) {            // Wv: [((h*16+c)*8+i)][q*8+j]
      int t = idx - 8192;
      int qj = t % 48;
      int rest = t / 48;
      int i = rest & 7; rest >>= 3;
      int c = rest & 15;
      int h = rest >> 4;
      int q = qj >> 3, j = qj & 7;
      float s = 0.f;
      if (q == 0) {
#pragma unroll
        for (int b = 0; b < 8; ++b)
          s += v0p[(h * 16 + c) * 8 + b] * rb[(b * 8 + i) * 8 + j];
      } else if (q <= 2) {
        int o = q - 1;
#pragma unroll
        for (int b = 0; b < 8; ++b)
          s += v1p[(h * 16 + c) * 8 + b] * vb1[((b * 2 + o) * 8 + i) * 8 + j];
      } else {
        int o = q - 3;
#pragma unroll
        for (int b = 0; b < 8; ++b)
          s += v2p[(h * 16 + c) * 8 + b] * vb2[((b * 3 + o) * 8 + i) * 8 + j];
      }
      ws[idx] = s;
    } else {                             // WMt[p][k], p=(o,i), k=(hc,j)
      int t = idx - 32768;
      int p = t >> 9, k = t & 511;
      int o = p >> 3, i = p & 7, hc = k >> 3, j = k & 7;
      float s = 0.f;
#pragma unroll
      for (int b = 0; b < 8; ++b)
        s += wm[(o * 64 + hc) * 8 + b] * rb[(b * 8 + i) * 8 + j];
      ws[idx] = s;
    }
  }
}

// ---------------------------------------------------------------------------
// Kernel 2: per-edge transport f = pt @ (mask * x[neighbor]), then
// K[16 edges, 32] = f[16,128] @ WK^T[128,32] via V_WMMA_F32_16X16X4_F32.
// Block = 64 threads (2 waves) handles 16 edges; each wave builds f for 8
// edges (VALU, 8x8 transports), then each wave owns one 16-wide N tile of
// the K GEMM (M=16 edges, K=128 -> 32 WMMAs/wave).
// LDS strides 132 == 4 (mod 64 banks) -> conflict-free b64 A/B reads.
// ---------------------------------------------------------------------------
#define FSTR 132
__global__ void k_edges(const float* __restrict__ x,
                        const int* __restrict__ nbr,
                        const unsigned char* __restrict__ mask,
                        const float* __restrict__ wk,
                        const float* __restrict__ pt,
                        float* __restrict__ kbuf) {
  __shared__ float xs[2][128];
  __shared__ float ps[2][64];
  __shared__ float fs[16][FSTR];
  __shared__ float wks[32][FSTR];
  int tid = threadIdx.x;
  int lane = tid & 31;
  int wid = tid >> 5;
  long e0 = (long)blockIdx.x * 16;

  // Stage WK [32][128] once per block
  for (int idx = tid; idx < 32 * 32; idx += 64) {
    int r = idx >> 5, c4 = idx & 31;
    flt4 val = *(const flt4*)(wk + r * 128 + c4 * 4);
    *(flt4*)&wks[r][c4 * 4] = val;
  }

  // f phase: wave `wid` handles edges e0 + wid*8 .. +8
  int c = lane >> 1;
  int dbase = (lane & 1) * 4;
  for (int it = 0; it < 8; ++it) {
    long e = e0 + wid * 8 + it;
    int nb = nbr[e];
    float m = mask[e] ? 1.f : 0.f;
    flt4 xv = *(const flt4*)(x + (long)nb * 128 + lane * 4);
    v2f pv = *(const v2f*)(pt + e * 64 + lane * 2);
    __syncthreads();  // previous iteration's reads done before overwrite
    xs[wid][lane * 4 + 0] = xv.x * m;
    xs[wid][lane * 4 + 1] = xv.y * m;
    xs[wid][lane * 4 + 2] = xv.z * m;
    xs[wid][lane * 4 + 3] = xv.w * m;
    ps[wid][lane * 2 + 0] = pv.x;
    ps[wid][lane * 2 + 1] = pv.y;
    __syncthreads();
#pragma unroll
    for (int t = 0; t < 4; ++t) {
      int d = dbase + t;
      float s = 0.f;
#pragma unroll
      for (int j = 0; j < 8; ++j) s += ps[wid][d * 8 + j] * xs[wid][c * 8 + j];
      fs[wid * 8 + it][c * 8 + dbase + t] = s;
    }
  }
  __syncthreads();  // fs + wks complete

  // WMMA phase: D[M=edge, N=hi] ; wave owns N tile [wid*16, wid*16+16)
  int arow = lane & 15;
  int koff = (lane >> 4) * 2;  // A VGPR0 holds K=0 (lanes 0-15) / K=2 (16-31)
  const float* ap = &fs[arow][koff];
  const float* bp = &wks[wid * 16 + arow][koff];
  v8f acc = {0.f, 0.f, 0.f, 0.f, 0.f, 0.f, 0.f, 0.f};
#pragma unroll 4
  for (int k0 = 0; k0 < 128; k0 += 4) {
    v2f a = *(const v2f*)(ap + k0);
    v2f b = *(const v2f*)(bp + k0);
    acc = __builtin_amdgcn_wmma_f32_16x16x4_f32(
        false, a, false, b, (short)0, acc, false, false);
  }
  int Mbase = (lane >> 4) * 8;  // D: VGPR r -> M=r (+8 for hi lanes)
  int Ncol = lane & 15;
#pragma unroll
  for (int r = 0; r < 8; ++r)
    kbuf[(e0 + Mbase + r) * 32 + wid * 16 + Ncol] = acc[r];
}

// ---------------------------------------------------------------------------
// Kernel 3: per-vertex attention. One wave per vertex; lane = (h,i).
// ---------------------------------------------------------------------------
__global__ void k_attn(const float* __restrict__ x,
                       const unsigned char* __restrict__ mask,
                       const float* __restrict__ wq,
                       const float* __restrict__ kbuf,
                       float* __restrict__ attn) {
  int lane = threadIdx.x & 31;
  int wid = threadIdx.x >> 5;
  long v = (long)blockIdx.x * 8 + wid;

  const float* row = wq + lane * 128;
  const float* xr = x + v * 128;
  float q = 0.f;
#pragma unroll
  for (int k4 = 0; k4 < 32; ++k4) {
    flt4 w4 = *(const flt4*)(row + k4 * 4);
    flt4 xv = *(const flt4*)(xr + k4 * 4);
    q += w4.x * xv.x + w4.y * xv.y + w4.z * xv.z + w4.w * xv.w;
  }

  float sc[NEI], mloc[NEI];
  float den = 0.f, nval = 0.f;
#pragma unroll
  for (int n = 0; n < NEI; ++n) {
    float kv = kbuf[(v * NEI + n) * 32 + lane];
    float t = q + kv;
    t = t > 0.f ? t : 0.f;
    t += __shfl_xor(t, 1, 32);   // reduce over i (groups of 8 lanes)
    t += __shfl_xor(t, 2, 32);
    t += __shfl_xor(t, 4, 32);
    float s = t * 0.125f;
    float m = mask[v * NEI + n] ? 1.f : 0.f;
    s *= m;
    sc[n] = s; mloc[n] = m;
    den += s; nval += m;
  }
  float nvalid = nval > 1.f ? nval : 1.f;
  int dead = den < 1e-6f;
  float invden = 1.f / (den > 1e-6f ? den : 1e-6f);
  if ((lane & 7) == 0) {
    int h = lane >> 3;
#pragma unroll
    for (int n = 0; n < NEI; ++n) {
      float a = dead ? (mloc[n] / nvalid) : sc[n] * invden;
      attn[(v * NEI + n) * 4 + h] = a;
    }
  }
}

// ---------------------------------------------------------------------------
// Kernel 4: per-vertex moments g[h][q][c][j] (registers) + head = Wv . g.
// One wave per vertex; 4 waves per block.
// ---------------------------------------------------------------------------
__global__ void k_head(const float* __restrict__ x,
                       const int* __restrict__ nbr,
                       const unsigned char* __restrict__ mask,
                       const float* __restrict__ pt,
                       const float* __restrict__ rpu,
                       const float* __restrict__ wv_,
                       const float* __restrict__ attn,
                       float* __restrict__ head) {
  __shared__ float xs[4][128];
  __shared__ float ps[4][64];
  __shared__ float gs[4][3072];
  int lane = threadIdx.x & 31;
  int wid = threadIdx.x >> 5;
  long v = (long)blockIdx.x * 4 + wid;

  float g[4][6][4];
#pragma unroll
  for (int h = 0; h < 4; ++h)
#pragma unroll
    for (int q = 0; q < 6; ++q)
#pragma unroll
      for (int t = 0; t < 4; ++t) g[h][q][t] = 0.f;

  int c = lane >> 1;
  int dbase = (lane & 1) * 4;

  for (int n = 0; n < NEI; ++n) {
    long e = v * NEI + n;
    int nb = nbr[e];
    float m = mask[e] ? 1.f : 0.f;
    flt4 xv = *(const flt4*)(x + (long)nb * 128 + lane * 4);
    v2f pv = *(const v2f*)(pt + e * 64 + lane * 2);
    __syncthreads();  // previous-iteration LDS reads done before overwrite
    xs[wid][lane * 4 + 0] = xv.x * m;
    xs[wid][lane * 4 + 1] = xv.y * m;
    xs[wid][lane * 4 + 2] = xv.z * m;
    xs[wid][lane * 4 + 3] = xv.w * m;
    ps[wid][lane * 2 + 0] = pv.x;
    ps[wid][lane * 2 + 1] = pv.y;
    __syncthreads();

    float fr[4];
#pragma unroll
    for (int t = 0; t < 4; ++t) {
      int d = dbase + t;
      float s = 0.f;
#pragma unroll
      for (int j = 0; j < 8; ++j) s += ps[wid][d * 8 + j] * xs[wid][c * 8 + j];
      fr[t] = s;
    }
    float u0 = rpu[e * 2 + 0], u1 = rpu[e * 2 + 1];
    float poly[6] = {1.f, u0, u1, u0 * u0, 2.f * u0 * u1, u1 * u1};
    float w[4][6];
#pragma unroll
    for (int h = 0; h < 4; ++h) {
      float ah = attn[e * 4 + h];
#pragma unroll
      for (int q = 0; q < 6; ++q) w[h][q] = ah * poly[q];
    }
#pragma unroll
    for (int h = 0; h < 4; ++h)
#pragma unroll
      for (int q = 0; q < 6; ++q)
#pragma unroll
        for (int t = 0; t < 4; ++t) g[h][q][t] += w[h][q] * fr[t];
  }

#pragma unroll
  for (int h = 0; h < 4; ++h)
#pragma unroll
    for (int q = 0; q < 6; ++q)
#pragma unroll
      for (int t = 0; t < 4; ++t)
        gs[wid][(h * 6 + q) * 128 + c * 8 + dbase + t] = g[h][q][t];
  __syncthreads();

  // head[v][o], o = (h*16+c)*8+i; 16 consecutive outputs per lane.
#pragma unroll
  for (int s = 0; s < 16; ++s) {
    int o = lane * 16 + s;
    int h = o >> 7, ci = o & 127, cc = ci >> 3, i = ci & 7;
    const float* wrow = wv_ + (long)((h * 16 + cc) * 8 + i) * 48;
    float acc = 0.f;
#pragma unroll
    for (int q = 0; q < 6; ++q)
#pragma unroll
      for (int j = 0; j < 8; ++j)
        acc += wrow[q * 8 + j] * gs[wid][(h * 6 + q) * 128 + cc * 8 + j];
    head[v * 512 + o] = acc;
  }
}

// ---------------------------------------------------------------------------
// Kernel 5: out[V,128] = head[V,512] @ WMt^T via V_WMMA_F32_16X16X4_F32.
// grid = (250, 4): y = 32-wide p slice; x covers 5 vertex tiles of 16 each.
// Block = 64 threads (2 waves), each wave owns a 16-wide N tile.
// LDS: wm_s[32][516] + a_s[16][516]  (stride 516 -> conflict-free b64 reads).
// ---------------------------------------------------------------------------
#define ASTR 516
__global__ void k_out(const float* __restrict__ wmt,
                      const float* __restrict__ head,
                      float* __restrict__ out) {
  extern __shared__ float lds[];
  float* wm_s = lds;                 // [32][ASTR]
  float* a_s = lds + 32 * ASTR;      // [16][ASTR]
  int tid = threadIdx.x;
  int lane = tid & 31;
  int wid = tid >> 5;
  int p0 = blockIdx.y * 32;

  // Stage WMt rows p0..p0+31 once per block (reused across 5 vertex tiles)
  for (int idx = tid; idx < 32 * 128; idx += 64) {
    int r = idx >> 7, c4 = idx & 127;
    flt4 val = *(const flt4*)(wmt + ((long)(p0 + r)) * 512 + c4 * 4);
    *(flt4*)&wm_s[r * ASTR + c4 * 4] = val;
  }

  int arow = lane & 15;
  int koff = (lane >> 4) * 2;  // K=0/1 for lanes 0-15, K=2/3 for lanes 16-31
  const float* ap = a_s + arow * ASTR + koff;
  const float* bp = wm_s + (wid * 16 + arow) * ASTR + koff;

  for (int tt = 0; tt < 5; ++tt) {
    long vbase = ((long)blockIdx.x * 5 + tt) * 16;
    __syncthreads();  // prior tile's compute done before a_s overwrite
    for (int idx = tid; idx < 16 * 128; idx += 64) {
      int r = idx >> 7, c4 = idx & 127;
      flt4 val = *(const flt4*)(head + (vbase + r) * 512 + c4 * 4);
      *(flt4*)&a_s[r * ASTR + c4 * 4] = val;
    }
    __syncthreads();

    v8f acc = {0.f, 0.f, 0.f, 0.f, 0.f, 0.f, 0.f, 0.f};
#pragma unroll 4
    for (int k0 = 0; k0 < 512; k0 += 4) {
      v2f a = *(const v2f*)(ap + k0);
      v2f b = *(const v2f*)(bp + k0);
      acc = __builtin_amdgcn_wmma_f32_16x16x4_f32(
          /*neg_a=*/false, a, /*neg_b=*/false, b,
          /*c_mod=*/(short)0, acc, /*reuse_a=*/false, /*reuse_b=*/false);
    }

    int Mbase = (lane >> 4) * 8;  // D layout: VGPR r -> M=r (+8 for hi lanes)
    int Ncol = lane & 15;
#pragma unroll
    for (int r = 0; r < 8; ++r)
      out[(vbase + Mbase + r) * 128 + p0 + wid * 16 + Ncol] = acc[r];
  }
}

// ---------------------------------------------------------------------------
extern "C" void kernel_launch(void* const* d_in, const int* in_sizes, int n_in,
                              void* d_out, int out_size, void* d_ws,
                              size_t ws_size, hipStream_t stream) {
  const float* x = (const float*)d_in[0];
  const int* nbr = (const int*)d_in[1];
  const unsigned char* mask = (const unsigned char*)d_in[2];
  const float* pt = (const float*)d_in[3];
  const float* rpu = (const float*)d_in[4];
  const float* rb = (const float*)d_in[5];
  const float* vb1 = (const float*)d_in[6];
  const float* vb2 = (const float*)d_in[7];
  const float* qc = (const float*)d_in[8];
  const float* kc = (const float*)d_in[9];
  const float* v0p = (const float*)d_in[10];
  const float* v1p = (const float*)d_in[11];
  const float* v2p = (const float*)d_in[12];
  const float* wm = (const float*)d_in[13];

  float* ws = (float*)d_ws;
  float* wq = ws + WS_WQ;
  float* wk = ws + WS_WK;
  float* wvv = ws + WS_WV;
  float* wmt = ws + WS_WMT;
  float* kbuf = ws + WS_KBUF;
  float* attn = ws + WS_ATTN;
  float* head = ws + WS_HEAD;
  float* out = (float*)d_out;

  k_weights<<<96, 256, 0, stream>>>(rb, vb1, vb2, qc, kc, v0p, v1p, v2p, wm, ws);
  k_edges<<<VNUM * NEI / 16, 64, 0, stream>>>(x, nbr, mask, wk, pt, kbuf);
  k_attn<<<VNUM / 8, 256, 0, stream>>>(x, mask, wq, kbuf, attn);
  k_head<<<VNUM / 4, 128, 0, stream>>>(x, nbr, mask, pt, rpu, wvv, attn, head);
  size_t ldsb = (size_t)(32 * ASTR + 16 * ASTR) * sizeof(float);  // 99072 B
  k_out<<<dim3(250, 4), 64, ldsb, stream>>>(wmt, head, out);
}